// GraphConstructor_63471026700851
// MI455X (gfx1250) — compile-verified
//
#include <hip/hip_runtime.h>
#include <hip/hip_bf16.h>

typedef __attribute__((ext_vector_type(2))) float v2f;
typedef __attribute__((ext_vector_type(8))) float v8f;

#define NN    2048
#define EDIM  32
#define R2    0.0025f   // 0.05^2

// One wave computes edge_attr[s, d_base:d_base+16, 0:32] via
// layer1 on VALU (K=2) + 16x V_WMMA_F32_16X16X4_F32 for layer2 (K=32).
__global__ __launch_bounds__(256) void GraphConstructor_wmma_kernel(
    const float* __restrict__ pos,   // [N,2]
    const float* __restrict__ W1,    // [2,32] row-major
    const float* __restrict__ b1,    // [32]
    const float* __restrict__ W2,    // [32,32] row-major
    const float* __restrict__ b2,    // [32]
    float* __restrict__ out)         // [N,N,32]
{
    const int lane = threadIdx.x & 31;
    const int wave = threadIdx.x >> 5;
    const int tile = blockIdx.x * 8 + wave;      // 0 .. 2048*128-1
    const int s      = tile >> 7;                // source node
    const int d_base = (tile & 127) << 4;        // 16 dst rows per tile

    const int half = lane >> 4;                  // 0: lanes 0-15, 1: lanes 16-31
    const int L    = lane & 15;

    // source position (wave-uniform)
    const float sx = pos[2 * s + 0];
    const float sy = pos[2 * s + 1];

    // this lane's A-matrix row (dst index)
    const int   d   = d_base + L;
    const float rpx = pos[2 * d + 0] - sx;
    const float rpy = pos[2 * d + 1] - sy;

    // ---- Layer 1: h = relu(rel_pos @ W1 + b1), produced in WMMA A layout ----
    // A 16x4 f32 layout: lane l holds M = l&15; VGPR v = A[M][2*(l>>4) + v].
    // Chain over K=32 as 8 groups of 4: group g covers columns 4g .. 4g+3.
    v2f A[8];
#pragma unroll
    for (int g = 0; g < 8; ++g) {
        const int e0 = 4 * g + 2 * half;
        float h0 = rpx * W1[e0]     + rpy * W1[32 + e0]     + b1[e0];
        float h1 = rpx * W1[e0 + 1] + rpy * W1[32 + e0 + 1] + b1[e0 + 1];
        A[g].x = h0 > 0.0f ? h0 : 0.0f;
        A[g].y = h1 > 0.0f ? h1 : 0.0f;
    }

    // ---- W2 in WMMA B layout (B 4x16 f32: lane l holds N = l&15,
    // VGPR v = B[2*(l>>4) + v][N]). Two N-tiles cover the 32 output dims.
    v2f B0[8], B1[8];
#pragma unroll
    for (int g = 0; g < 8; ++g) {
        const int k = 4 * g + 2 * half;
        B0[g].x = W2[(k + 0) * 32 + L];
        B0[g].y = W2[(k + 1) * 32 + L];
        B1[g].x = W2[(k + 0) * 32 + 16 + L];
        B1[g].y = W2[(k + 1) * 32 + 16 + L];
    }

    // ---- C init with bias: C layout comp v = (row v + 8*half, col L) -> b2[col]
    v8f acc0, acc1;
    const float bz0 = b2[L];
    const float bz1 = b2[16 + L];
#pragma unroll
    for (int v = 0; v < 8; ++v) { acc0[v] = bz0; acc1[v] = bz1; }

    // ---- Layer 2: 8 chained K=4 fp32 WMMAs per 16-column accumulator ----
#pragma unroll
    for (int g = 0; g < 8; ++g) {
        acc0 = __builtin_amdgcn_wmma_f32_16x16x4_f32(
            false, A[g], false, B0[g], (short)0, acc0, false, false);
        acc1 = __builtin_amdgcn_wmma_f32_16x16x4_f32(
            false, A[g], false, B1[g], (short)0, acc1, false, false);
    }

    // ---- Radius mask + non-temporal stores (output >> L2, don't pollute) ----
    float* outS = out + ((size_t)s * NN + (size_t)d_base) * EDIM;
#pragma unroll
    for (int v = 0; v < 8; ++v) {
        const int   row = v + 8 * half;
        const int   dr  = d_base + row;
        const float dx  = pos[2 * dr + 0] - sx;
        const float dy  = pos[2 * dr + 1] - sy;
        const bool  keep = (dx * dx + dy * dy <= R2) && (dr != s);
        const float m   = keep ? 1.0f : 0.0f;
        float* p = outS + (size_t)row * EDIM + L;
        __builtin_nontemporal_store(m * acc0[v], p);
        __builtin_nontemporal_store(m * acc1[v], p + 16);
    }
}

extern "C" void kernel_launch(void* const* d_in, const int* in_sizes, int n_in,
                              void* d_out, int out_size, void* d_ws, size_t ws_size,
                              hipStream_t stream) {
    // setup_inputs order:
    //   0: node_features [N,128] (pass-through, unused in output)
    //   1: object_positions_2d [N,2]
    //   2: W1 [2,32]  3: b1 [32]  4: W2 [32,32]  5: b2 [32]
    const float* pos = (const float*)d_in[1];
    const float* W1  = (const float*)d_in[2];
    const float* b1  = (const float*)d_in[3];
    const float* W2  = (const float*)d_in[4];
    const float* b2  = (const float*)d_in[5];
    float* out = (float*)d_out;

    // 2048 sources * 128 dst-tiles = 262144 waves; 8 waves per 256-thread block
    const int num_tiles  = NN * (NN / 16);
    const int num_blocks = num_tiles / 8;   // 32768
    GraphConstructor_wmma_kernel<<<num_blocks, 256, 0, stream>>>(
        pos, W1, b1, W2, b2, out);
}